// FrangiLoss_70970039599593
// MI455X (gfx1250) — compile-verified
//
#include <hip/hip_runtime.h>
#include <math.h>
#include <stdint.h>

#define DD 64
#define HH 128
#define WW 128
#define NVOX (DD*HH*WW)   /* 1048576 = 2^20 ; HH*WW = 16384 = 2^14 */

typedef __attribute__((ext_vector_type(2))) float v2f;
typedef __attribute__((ext_vector_type(8))) float v8f;
typedef __attribute__((ext_vector_type(4))) unsigned int v4u;
typedef __attribute__((ext_vector_type(4))) int v4i;
typedef __attribute__((ext_vector_type(8))) int v8i;

#if __has_builtin(__builtin_amdgcn_tensor_load_to_lds) && __has_builtin(__builtin_amdgcn_s_wait_tensorcnt)
#define HAVE_TDM 1
#else
#define HAVE_TDM 0
#endif

struct Taps { float g[3][3]; float d1[3][3]; float d2[3][3]; float s2[3]; };

__device__ __forceinline__ int refl(int i, int n) {
  i = (i < 0) ? -i : i;
  return (i >= n) ? (2*n - 2 - i) : i;
}

__device__ __forceinline__ float blockSum(float v, float* red) {
  int tid = threadIdx.x;
  red[tid] = v;
  __syncthreads();
  for (int s = 128; s > 0; s >>= 1) {
    if (tid < s) red[tid] += red[tid + s];
    __syncthreads();
  }
  float r = red[0];
  __syncthreads();
  return r;
}

// ---------------- Kernel A: softmax(ch1) + background-loss partials ----------------
__global__ void kSoftmaxBg(const float* __restrict__ net, const float* __restrict__ data,
                           float* __restrict__ vein, float* __restrict__ pA) {
  __shared__ float red[256];
  int tid = threadIdx.x;
  float bgn = 0.f, bgd = 0.f, cnt = 0.f;
  for (long long idx = (long long)blockIdx.x * 256 + tid; idx < 2LL * NVOX; idx += 262144LL) {
    int b = (int)(idx >> 20);
    int i = (int)(idx & (NVOX - 1));
    float x0 = net[((size_t)(b*2+0) << 20) + i];
    float x1 = net[((size_t)(b*2+1) << 20) + i];
    float p = 1.f / (1.f + expf(x0 - x1));      // softmax over 2 channels
    vein[idx] = p;
    float chi = data[((size_t)(b*3+0) << 20) + i];
    float vi  = data[((size_t)(b*3+2) << 20) + i];
    float nv = (vi < 0.05f) ? 1.f : 0.f;
    bgn += p * nv;
    bgd += nv;
    cnt += ((p >= 0.5f) && (chi != 0.f)) ? 1.f : 0.f;
  }
  float s0 = blockSum(bgn, red);
  float s1 = blockSum(bgd, red);
  float s2 = blockSum(cnt, red);
  if (tid == 0) { pA[blockIdx.x*3+0] = s0; pA[blockIdx.x*3+1] = s1; pA[blockIdx.x*3+2] = s2; }
}

// ---------------- Kernel B: V_gate = sigmoid(6*(maxpool3(V_I) - 1e-5)) ----------------
// (sigmoid is monotone -> maxpool(sigmoid(x)) == sigmoid(maxpool(x)))
__global__ void kGate(const float* __restrict__ data, float* __restrict__ gate) {
  long long idx = (long long)blockIdx.x * 256 + threadIdx.x;
  if (idx >= 2LL * NVOX) return;
  int b = (int)(idx >> 20);
  int i = (int)(idx & (NVOX - 1));
  int z = i >> 14;
  int y = (i >> 7) & 127;
  int x = i & 127;
  const float* vi = data + ((size_t)(b*3+2) << 20);
  float m = -1e30f;
  for (int dz = -1; dz <= 1; ++dz) {
    int zz = z + dz; if (zz < 0 || zz >= DD) continue;
    for (int dy = -1; dy <= 1; ++dy) {
      int yy = y + dy; if (yy < 0 || yy >= HH) continue;
      for (int dx = -1; dx <= 1; ++dx) {
        int xx = x + dx; if (xx < 0 || xx >= WW) continue;
        m = fmaxf(m, vi[((size_t)zz << 14) + (yy << 7) + xx]);
      }
    }
  }
  gate[idx] = 1.f / (1.f + expf(-6.f * (m - 1e-5f)));
}

// ---------------- Kernel C: P_blur via banded WMMA separable conv ----------------
// One wave per 16x16 (y,x) tile of one z-slice. z-conv (VALU sum, reflect) -> LDS t[18][18],
// y-conv: D = Band_y(16x20) x t(20x16) via 5x V_WMMA_F32_16X16X4_F32,
// x-conv: D = u(16x20) x Band_x(20x16)/27 via 5x WMMA.
__global__ void kBlur(const float* __restrict__ vein, float* __restrict__ blur) {
  __shared__ float tb[8][324];   // per-wave 18x18
  __shared__ float ub[8][288];   // per-wave 16x18
  int tid = threadIdx.x, w = tid >> 5, lane = tid & 31;
  int tileId = blockIdx.x * 8 + w;                 // 8192 tiles = b(2)*z(64)*yt(8)*xt(8)
  int xt = tileId & 7, yt = (tileId >> 3) & 7, z = (tileId >> 6) & 63, b = tileId >> 12;
  int x0 = xt * 16, y0 = yt * 16;
  float* t = tb[w];
  float* u = ub[w];
  const float* src = vein + ((size_t)b << 20);
  int zm = refl(z - 1, DD), zp = refl(z + 1, DD);
  for (int k = lane; k < 324; k += 32) {
    int iy = k / 18, ix = k - iy * 18;
    int gy = refl(y0 - 1 + iy, HH), gx = refl(x0 - 1 + ix, WW);
    size_t o = ((size_t)gy << 7) + gx;
    t[k] = src[((size_t)zm << 14) + o] + src[((size_t)z << 14) + o] + src[((size_t)zp << 14) + o];
  }
  __syncthreads();
  int q = lane & 15;       // A: output row ; B: output column
  bool hi = lane >= 16;
  // ---- y conv: out u[r][c] = sum_j Band[r][j]*t[j][c], Band[r][j]=1 for j-r in [0,2]
  v8f acc = {0.f,0.f,0.f,0.f,0.f,0.f,0.f,0.f};
#pragma unroll
  for (int kc = 0; kc < 5; kc++) {
    int k0 = 4*kc + (hi ? 2 : 0), k1 = k0 + 1;
    v2f a, bv;
    a.x = (k0 >= q && k0 <= q+2 && k0 < 18) ? 1.f : 0.f;
    a.y = (k1 >= q && k1 <= q+2 && k1 < 18) ? 1.f : 0.f;
    int j0 = (k0 < 18) ? k0 : 17, j1 = (k1 < 18) ? k1 : 17;
    bv.x = t[j0*18 + q];
    bv.y = t[j1*18 + q];
    acc = __builtin_amdgcn_wmma_f32_16x16x4_f32(false, a, false, bv, (short)0, acc, false, false);
  }
#pragma unroll
  for (int v = 0; v < 8; v++) {
    int r = hi ? (v + 8) : v;
    u[r*18 + q] = acc[v];
  }
  { // halo columns 16,17 by VALU (one value per lane)
    int c = hi ? 17 : 16;
    u[q*18 + c] = t[q*18 + c] + t[(q+1)*18 + c] + t[(q+2)*18 + c];
  }
  __syncthreads();
  // ---- x conv: out[r][c] = sum_j u[r][j]*Band[j][c]/27, Band[j][c]=1 for j-c in [0,2]
  v8f acc2 = {0.f,0.f,0.f,0.f,0.f,0.f,0.f,0.f};
  const float inv27 = 1.f / 27.f;
#pragma unroll
  for (int kc = 0; kc < 5; kc++) {
    int k0 = 4*kc + (hi ? 2 : 0), k1 = k0 + 1;
    v2f a, bv;
    a.x = (k0 < 18) ? u[q*18 + k0] : 0.f;
    a.y = (k1 < 18) ? u[q*18 + k1] : 0.f;
    bv.x = (k0 >= q && k0 <= q+2 && k0 < 18) ? inv27 : 0.f;
    bv.y = (k1 >= q && k1 <= q+2 && k1 < 18) ? inv27 : 0.f;
    acc2 = __builtin_amdgcn_wmma_f32_16x16x4_f32(false, a, false, bv, (short)0, acc2, false, false);
  }
  float* dst = blur + ((size_t)b << 20) + ((size_t)z << 14);
#pragma unroll
  for (int v = 0; v < 8; v++) {
    int r = hi ? (v + 8) : v;
    dst[((size_t)(y0 + r) << 7) + (x0 + q)] = acc2[v];
  }
}

// ---------------- slice loader for Frangi: TDM for interior tiles, VALU+reflect else ----
__device__ __forceinline__ void loadSlice(float* Ss, const float* src, int zz,
                                          int y0, int x0, bool interior, int tid) {
#if HAVE_TDM
  if (interior) {
    if (tid < 32) {
      unsigned long long ga = (unsigned long long)(uintptr_t)
          (src + ((size_t)zz << 14) + ((size_t)(y0 - 1) << 7) + (x0 - 1));
      unsigned ldsOff = (unsigned)(uintptr_t)Ss;
      v4u g0;
      g0[0] = 1u;                                               // count=1
      g0[1] = ldsOff;                                           // lds_addr
      g0[2] = (unsigned)ga;                                     // global_addr[31:0]
      g0[3] = ((unsigned)(ga >> 32) & 0x01FFFFFFu) | 0x80000000u; // addr[56:32] | type=2
      v8i g1;
      g1[0] = 0x00020000;        // data_size = 4B
      g1[1] = (int)(18u << 16);  // tensor_dim0 = 18
      g1[2] = (int)(18u << 16);  // tensor_dim1 = 18
      g1[3] = (int)(18u << 16);  // tile_dim0 = 18
      g1[4] = 18;                // tile_dim1 = 18, tile_dim2 = 0 (2D)
      g1[5] = WW;                // tensor_dim0_stride = 128
      g1[6] = 0;
      g1[7] = 0;
      v4i g2 = {0,0,0,0};
      v4i g3 = {0,0,0,0};
#if defined(__clang_major__) && __clang_major__ >= 23
      v8i g4 = {0,0,0,0,0,0,0,0};
      __builtin_amdgcn_tensor_load_to_lds(g0, g1, g2, g3, g4, 0);
#else
      __builtin_amdgcn_tensor_load_to_lds(g0, g1, g2, g3, 0);
#endif
      __builtin_amdgcn_s_wait_tensorcnt(0);
    }
    return;
  }
#endif
  (void)interior;
  for (int k = tid; k < 324; k += 256) {
    int iy = k / 18, ix = k - iy * 18;
    int gy = refl(y0 - 1 + iy, HH), gx = refl(x0 - 1 + ix, WW);
    Ss[k] = src[((size_t)zz << 14) + ((size_t)gy << 7) + gx];
  }
}

__device__ __forceinline__ void eig3abs(float a00, float a01, float a02,
                                        float a11, float a12, float a22,
                                        float& l1, float& l2, float& l3) {
  float p1 = a01*a01 + a02*a02 + a12*a12;
  float q  = (a00 + a11 + a22) * (1.f/3.f);
  float b00 = a00 - q, b11 = a11 - q, b22 = a22 - q;
  float p2 = b00*b00 + b11*b11 + b22*b22 + 2.f*p1;
  float p  = sqrtf(p2 * (1.f/6.f) + 1e-32f);
  float ip = 1.f / p;
  float c00 = b00*ip, c01 = a01*ip, c02 = a02*ip, c11 = b11*ip, c12 = a12*ip, c22 = b22*ip;
  float detB = c00*(c11*c22 - c12*c12) - c01*(c01*c22 - c12*c02) + c02*(c01*c12 - c11*c02);
  float r = fminf(1.f, fmaxf(-1.f, 0.5f*detB));
  float phi = acosf(r) * (1.f/3.f);
  float eBig = q + 2.f*p*cosf(phi);
  float eSml = q + 2.f*p*cosf(phi + 2.0943951023931953f);
  float eMid = 3.f*q - eBig - eSml;
  float a = eBig, b = eMid, c = eSml, t;
  if (fabsf(a) > fabsf(b)) { t=a; a=b; b=t; }
  if (fabsf(b) > fabsf(c)) { t=b; b=c; c=t; }
  if (fabsf(a) > fabsf(b)) { t=a; a=b; b=t; }
  l1 = a; l2 = b; l3 = c;
}

// ---------------- Kernel D: Frangi vesselness + selfV/hinge partials ----------------
__global__ void kFrangi(const float* __restrict__ blur, const float* __restrict__ gate,
                        const float* __restrict__ vein, const float* __restrict__ data,
                        Taps tp, float* __restrict__ pD) {
  __shared__ float S[3][324];
  __shared__ float red[256];
  int tid = threadIdx.x;
  int ty = tid >> 4, tx = tid & 15;
  int bid = blockIdx.x;                    // 512 = b(2)*zc(4)*yt(8)*xt(8)
  int xt = bid & 7, yt = (bid >> 3) & 7, zc = (bid >> 6) & 3, b = bid >> 8;
  int x0 = xt*16, y0 = yt*16, z0 = zc*16;
  const float* src = blur + ((size_t)b << 20);
  bool interior = (y0 > 0) && (y0 + 17 <= HH) && (x0 > 0) && (x0 + 17 <= WW);
  loadSlice(S[0], src, refl(z0 - 1, DD), y0, x0, interior, tid);
  loadSlice(S[1], src, z0, y0, x0, interior, tid);
  float sumSelf = 0.f, sumHinge = 0.f;
  for (int zi = 0; zi < 16; ++zi) {
    int z = z0 + zi;
    loadSlice(S[(zi + 2) % 3], src, refl(z + 1, DD), y0, x0, interior, tid);
    __syncthreads();
    const float* Sp = S[zi % 3];
    const float* Sc = S[(zi + 1) % 3];
    const float* Sn = S[(zi + 2) % 3];
    float nb[3][3][3];
#pragma unroll
    for (int dy = 0; dy < 3; dy++)
#pragma unroll
      for (int dx = 0; dx < 3; dx++) {
        int o = (ty + dy)*18 + tx + dx;
        nb[0][dy][dx] = Sp[o]; nb[1][dy][dx] = Sc[o]; nb[2][dy][dx] = Sn[o];
      }
    float Vmax = 0.f;
#pragma unroll
    for (int s = 0; s < 3; s++) {
      float xg[3][3], xd1[3][3], xd2[3][3];
#pragma unroll
      for (int dz = 0; dz < 3; dz++)
#pragma unroll
        for (int dy = 0; dy < 3; dy++) {
          float n0 = nb[dz][dy][0], n1 = nb[dz][dy][1], n2 = nb[dz][dy][2];
          xg [dz][dy] = tp.g [s][0]*n0 + tp.g [s][1]*n1 + tp.g [s][2]*n2;
          xd1[dz][dy] = tp.d1[s][0]*n0 + tp.d1[s][1]*n1 + tp.d1[s][2]*n2;
          xd2[dz][dy] = tp.d2[s][0]*n0 + tp.d2[s][1]*n1 + tp.d2[s][2]*n2;
        }
      float Ixx=0.f, Iyy=0.f, Izz=0.f, Ixy=0.f, Ixz=0.f, Iyz=0.f;
#pragma unroll
      for (int dz = 0; dz < 3; dz++) {
        float txx=0.f, tyy=0.f, tzz=0.f, txy=0.f, txz=0.f, tyz=0.f;
#pragma unroll
        for (int dy = 0; dy < 3; dy++) {
          txx += tp.g [s][dy]*xd2[dz][dy];
          tyy += tp.d2[s][dy]*xg [dz][dy];
          tzz += tp.g [s][dy]*xg [dz][dy];
          txy += tp.d1[s][dy]*xd1[dz][dy];
          txz += tp.g [s][dy]*xd1[dz][dy];
          tyz += tp.d1[s][dy]*xg [dz][dy];
        }
        Ixx += tp.g [s][dz]*txx;
        Iyy += tp.g [s][dz]*tyy;
        Izz += tp.d2[s][dz]*tzz;
        Ixy += tp.g [s][dz]*txy;
        Ixz += tp.d1[s][dz]*txz;
        Iyz += tp.d1[s][dz]*tyz;
      }
      float sc = tp.s2[s];
      float l1, l2, l3;
      eig3abs(sc*Ixx, sc*Ixy, sc*Ixz, sc*Iyy, sc*Iyz, sc*Izz, l1, l2, l3);
      float Ra = fabsf(l2) / (fabsf(l3) + 1e-12f);
      float Rb = fabsf(l1) / sqrtf(fabsf(l2)*fabsf(l3) + 1e-12f);
      float S2 = fmaxf(l1*l1 + l2*l2 + l3*l3, 1e-12f);
      float V = expf(-Ra*Ra*(1.f/1.28f)) * expf(-Rb*Rb*(1.f/1.28f)) * (1.f - expf(-S2*(1.f/32.f)));
      V = ((l2 < 0.f) && (l3 < 0.f)) ? V : 0.f;
      V = fminf(fmaxf(V, 0.f), 1.f);
      Vmax = fmaxf(Vmax, V);
    }
    size_t off = ((size_t)z << 14) + ((size_t)(y0 + ty) << 7) + (x0 + tx);
    size_t gi = ((size_t)b << 20) + off;
    float g8  = gate[gi];
    float p   = vein[gi];
    float chi = data[((size_t)(b*3) << 20) + off];
    float vi  = data[((size_t)(b*3+2) << 20) + off];
    float valid = ((p >= 0.5f) && (chi != 0.f)) ? 1.f : 0.f;
    float gv = g8 * valid;
    sumSelf  += Vmax * gv;
    sumHinge += fmaxf(vi + 0.1f - Vmax, 0.f) * gv;
    __syncthreads();
  }
  float r0 = blockSum(sumSelf, red);
  float r1 = blockSum(sumHinge, red);
  if (tid == 0) { pD[bid*2] = r0; pD[bid*2+1] = r1; }
}

// ---------------- Kernel E: deterministic finalize ----------------
__global__ void kFinalize(const float* __restrict__ pA, const float* __restrict__ pD,
                          float* __restrict__ out) {
  if (threadIdx.x != 0 || blockIdx.x != 0) return;
  float bgn = 0.f, bgd = 0.f, cnt = 0.f, sv = 0.f, hg = 0.f;
  for (int i = 0; i < 1024; i++) { bgn += pA[3*i]; bgd += pA[3*i+1]; cnt += pA[3*i+2]; }
  for (int i = 0; i < 512; i++)  { sv += pD[2*i]; hg += pD[2*i+1]; }
  cnt = fmaxf(cnt, 1.f);
  bgd = fmaxf(bgd, 1.f);
  out[0] = -5.f*sv/cnt + 10.f*hg/cnt + 10.f*bgn/bgd;
}

extern "C" void kernel_launch(void* const* d_in, const int* in_sizes, int n_in,
                              void* d_out, int out_size, void* d_ws, size_t ws_size,
                              hipStream_t stream) {
  (void)in_sizes; (void)n_in; (void)out_size; (void)ws_size;
  const float* net  = (const float*)d_in[0];
  const float* data = (const float*)d_in[1];
  float* ws   = (float*)d_ws;
  float* vein = ws;
  float* blur = ws + 2*(size_t)NVOX;
  float* gate = ws + 4*(size_t)NVOX;
  float* pA   = ws + 6*(size_t)NVOX;
  float* pD   = pA + 1024*3;

  Taps tp;
  const float sig[3] = {0.01f, 0.2f, 0.3f};
  for (int s = 0; s < 3; s++) {
    float sigma = sig[s];
    float xs[3] = {-1.f, 0.f, 1.f};
    float gg[3]; float sum = 0.f;
    for (int i = 0; i < 3; i++) { float e = xs[i]/sigma; gg[i] = expf(-0.5f*e*e); sum += gg[i]; }
    for (int i = 0; i < 3; i++) gg[i] /= sum;
    float s2 = sigma*sigma;
    float t1[3]; float m1 = 0.f;
    for (int i = 0; i < 3; i++) { t1[i] = -(xs[i]/s2)*gg[i]; m1 += t1[i]; }
    m1 /= 3.f;
    float t2[3]; float m2 = 0.f;
    for (int i = 0; i < 3; i++) { t2[i] = (xs[i]*xs[i]/(s2*s2) - 1.f/s2)*gg[i]; m2 += t2[i]; }
    m2 /= 3.f;
    for (int i = 0; i < 3; i++) {
      tp.g[s][i]  = gg[i];
      tp.d1[s][i] = t1[i] - m1;
      tp.d2[s][i] = t2[i] - m2;
    }
    tp.s2[s] = s2;
  }

  kSoftmaxBg<<<1024, 256, 0, stream>>>(net, data, vein, pA);
  kGate<<<8192, 256, 0, stream>>>(data, gate);
  kBlur<<<1024, 256, 0, stream>>>(vein, blur);
  kFrangi<<<512, 256, 0, stream>>>(blur, gate, vein, data, tp, pD);
  kFinalize<<<1, 64, 0, stream>>>(pA, pD, (float*)d_out);
}